// VoxelNeRF_43465069035940
// MI455X (gfx1250) — compile-verified
//
#include <hip/hip_runtime.h>
#include <math.h>

#define M_TOTAL  4194304          // 16384 * 256
#define NRAYS    16384
#define BT       256              // threads per block (8 waves of 32)
#define WAVESZ   32
#define NWAVES   8
#define TILE     256              // elements per wave tile (16x16 matrix)
#define CHUNK    2048             // elements per block (8 tiles)
#define NBLK     (M_TOTAL / CHUNK) // 2048
#define IT       8                // elements per thread

typedef float v2f __attribute__((ext_vector_type(2)));
typedef float v8f __attribute__((ext_vector_type(8)));

// v = log(1 - alpha) = -softplus(d + shift) * interval   (exact, stable, <= 0)
__device__ __forceinline__ float log1ma(float d, float shift, float interval) {
  float x = d + shift;
  float sp = (x > 20.0f) ? x : log1pf(expf(x));
  return -sp * interval;
}

// ---------------------------------------------------------------------------
// K1: per-block segmented-sum carry: (anyFlag, sum of v after last ray head)
// ---------------------------------------------------------------------------
__global__ void k_blockcarry(const float* __restrict__ density,
                             const int*   __restrict__ ray_id,
                             const float* __restrict__ shiftp,
                             const float* __restrict__ intervalp,
                             float* __restrict__ carrySum,
                             int*   __restrict__ carryFlag) {
  __shared__ float s_ws[NWAVES];
  __shared__ int   s_wf[NWAVES];
  const float shift = shiftp[0], interval = intervalp[0];
  const int tid = threadIdx.x, lane = tid & (WAVESZ - 1), wave = tid >> 5;
  const int base = blockIdx.x * CHUNK + tid * IT;

  // vectorized contiguous loads: each thread owns 8 consecutive elements
  const float4* d4 = (const float4*)(density + base);
  const int4*   r4 = (const int4*)(ray_id + base);
  float dv[IT]; int rv[IT];
  float4 da = d4[0], db = d4[1];
  int4   ra = r4[0], rb = r4[1];
  dv[0]=da.x; dv[1]=da.y; dv[2]=da.z; dv[3]=da.w;
  dv[4]=db.x; dv[5]=db.y; dv[6]=db.z; dv[7]=db.w;
  rv[0]=ra.x; rv[1]=ra.y; rv[2]=ra.z; rv[3]=ra.w;
  rv[4]=rb.x; rv[5]=rb.y; rv[6]=rb.z; rv[7]=rb.w;

  int prev = (base == 0) ? (rv[0] - 1) : ray_id[base - 1];
  int f = 0; float s = 0.f;
  #pragma unroll
  for (int j = 0; j < IT; ++j) {
    float v = log1ma(dv[j], shift, interval);
    if (rv[j] != prev) { f = 1; s = v; } else { s += v; }
    prev = rv[j];
  }
  // wave32 inclusive scan of segmented-sum monoid
  #pragma unroll
  for (int off = 1; off < WAVESZ; off <<= 1) {
    int   of = __shfl_up(f, off);
    float os = __shfl_up(s, off);
    if (lane >= off) { s = f ? s : (os + s); f |= of; }
  }
  if (lane == WAVESZ - 1) { s_ws[wave] = s; s_wf[wave] = f; }
  __syncthreads();
  if (tid == 0) {
    int F = 0; float S = 0.f;
    for (int w = 0; w < NWAVES; ++w) {
      S = s_wf[w] ? s_ws[w] : (S + s_ws[w]);
      F |= s_wf[w];
    }
    carrySum[blockIdx.x] = S;
    carryFlag[blockIdx.x] = F;
  }
}

// ---------------------------------------------------------------------------
// K2: exclusive segmented-monoid scan over the NBLK block carries (1 block)
// ---------------------------------------------------------------------------
__global__ void k_scancarries(const float* __restrict__ carrySum,
                              const int*   __restrict__ carryFlag,
                              float* __restrict__ carryIn) {
  __shared__ float s_ws[NWAVES], s_we[NWAVES];
  __shared__ int   s_wf[NWAVES], s_wef[NWAVES];
  const int tid = threadIdx.x, lane = tid & (WAVESZ - 1), wave = tid >> 5;
  const int base = tid * IT;  // 256 * 8 == NBLK exactly

  float sv[IT]; int fv[IT];
  float ts = 0.f; int tf = 0;
  #pragma unroll
  for (int j = 0; j < IT; ++j) {
    sv[j] = carrySum[base + j];
    fv[j] = carryFlag[base + j];
    ts = fv[j] ? sv[j] : (ts + sv[j]);
    tf |= fv[j];
  }
  float is = ts; int ifl = tf;
  #pragma unroll
  for (int off = 1; off < WAVESZ; off <<= 1) {
    int   of = __shfl_up(ifl, off);
    float os = __shfl_up(is, off);
    if (lane >= off) { is = ifl ? is : (os + is); ifl |= of; }
  }
  float es = __shfl_up(is, 1); int ef = __shfl_up(ifl, 1);
  if (lane == 0) { es = 0.f; ef = 0; }
  if (lane == WAVESZ - 1) { s_ws[wave] = is; s_wf[wave] = ifl; }
  __syncthreads();
  if (tid == 0) {
    float S = 0.f; int F = 0;
    for (int w = 0; w < NWAVES; ++w) {
      s_we[w] = S; s_wef[w] = F;
      S = s_wf[w] ? s_ws[w] : (S + s_ws[w]);
      F |= s_wf[w];
    }
  }
  __syncthreads();
  float run = ef ? es : (s_we[wave] + es);   // combine(waveExcl, laneExcl).sum
  #pragma unroll
  for (int j = 0; j < IT; ++j) {
    carryIn[base + j] = run;                 // exclusive carry-in for block
    run = fv[j] ? sv[j] : (run + sv[j]);
  }
}

// ---------------------------------------------------------------------------
// K0: empty rays -> alphainv_last = 1.0
// ---------------------------------------------------------------------------
__global__ void k_init(float* __restrict__ out_a) {
  int i = blockIdx.x * blockDim.x + threadIdx.x;
  if (i < NRAYS) out_a[i] = 1.0f;
}

// ---------------------------------------------------------------------------
// K3: main pass. WMMA f32 tile prefix (P = X*U + Lstrict*R), segmented fixup,
//     weights = exp(S - v) * (-expm1(v)); segment ends write exp(S).
// ---------------------------------------------------------------------------
__global__ void k_main(const float* __restrict__ density,
                       const int*   __restrict__ ray_id,
                       const float* __restrict__ shiftp,
                       const float* __restrict__ intervalp,
                       const float* __restrict__ carryIn,
                       float* __restrict__ out_w,
                       float* __restrict__ out_a) {
  __shared__ float s_v[CHUNK];
  __shared__ float s_p[CHUNK];
  __shared__ unsigned char s_f[CHUNK];
  __shared__ float s_rsum[NWAVES][16];
  __shared__ float s_tsum[NWAVES];
  __shared__ int   s_tany[NWAVES];
  __shared__ float s_ct[NWAVES];

  const float shift = shiftp[0], interval = intervalp[0];
  const int tid = threadIdx.x, lane = tid & (WAVESZ - 1), wave = tid >> 5;
  const int cbase = blockIdx.x * CHUNK;

  // 1. coalesced: compute v = log(1-alpha) and ray-head flags into LDS
  for (int k = tid; k < CHUNK; k += BT) {
    int gi = cbase + k;
    int rid = ray_id[gi];
    s_v[k] = log1ma(density[gi], shift, interval);
    s_f[k] = (unsigned char)((gi == 0) || (ray_id[gi - 1] != rid));
  }
  __syncthreads();

  // 2. WMMA 256-element inclusive prefix per wave tile.
  //    X[r][c] = xt[16r + c]; P = X*U (row prefixes) + Lstrict*R (row offsets)
  const int half = lane >> 4;       // A/B reg layout: reg r, half h -> K = r + 2h
  const int l15  = lane & 15;
  const float* xt = &s_v[wave * TILE];
  v8f P = {};
  #pragma unroll
  for (int k0 = 0; k0 < 16; k0 += 4) {            // C1 = X * U
    int ka = k0 + 2 * half;
    v2f a, b;
    a.x = xt[l15 * 16 + ka];                      // A = X, M = l15, K = ka..ka+1
    a.y = xt[l15 * 16 + ka + 1];
    b.x = (ka     <= l15) ? 1.0f : 0.0f;          // B = U rows ka..ka+1, col l15
    b.y = (ka + 1 <= l15) ? 1.0f : 0.0f;
    P = __builtin_amdgcn_wmma_f32_16x16x4_f32(false, a, false, b, (short)0, P,
                                              false, false);
  }
  if (l15 == 15) {                                // col 15 of X*U = row sums
    #pragma unroll
    for (int g = 0; g < 8; ++g) s_rsum[wave][g + 8 * half] = P[g];
  }
  __syncthreads();
  #pragma unroll
  for (int k0 = 0; k0 < 16; k0 += 4) {            // P += Lstrict * R
    int ka = k0 + 2 * half;
    v2f a, b;
    a.x = (ka     < l15) ? 1.0f : 0.0f;           // A = Lstrict, M = l15
    a.y = (ka + 1 < l15) ? 1.0f : 0.0f;
    b.x = s_rsum[wave][ka];                       // B[k][c] = rowsum(k)
    b.y = s_rsum[wave][ka + 1];
    P = __builtin_amdgcn_wmma_f32_16x16x4_f32(false, a, false, b, (short)0, P,
                                              false, false);
  }
  #pragma unroll
  for (int g = 0; g < 8; ++g)                     // C/D layout: row g+8h, col l15
    s_p[wave * TILE + (g + 8 * half) * 16 + l15] = P[g];
  __syncthreads();

  // 3. propagate (seen-head?, plain-exclusive-prefix at latest head) per tile
  const int ebase = wave * TILE + lane * IT;      // 8 consecutive elems/thread
  int any = 0; float pex = 0.f;
  #pragma unroll
  for (int j = 0; j < IT; ++j) {
    int k = ebase + j;
    if (s_f[k]) { any = 1; pex = s_p[k] - s_v[k]; }
  }
  int ia = any; float ip = pex;
  #pragma unroll
  for (int off = 1; off < WAVESZ; off <<= 1) {
    int   oa = __shfl_up(ia, off);
    float op = __shfl_up(ip, off);
    if (lane >= off && !ia) { ia = oa; ip = op; } // combine: latest head wins
  }
  int ea = __shfl_up(ia, 1); float ep = __shfl_up(ip, 1);
  if (lane == 0) { ea = 0; ep = 0.f; }
  if (lane == WAVESZ - 1) {                       // tile carry for next tiles
    float Pend = s_p[wave * TILE + TILE - 1];
    s_tany[wave] = ia;
    s_tsum[wave] = ia ? (Pend - ip) : Pend;
  }
  __syncthreads();
  if (tid == 0) {                                 // chain carries across tiles
    float run = carryIn[blockIdx.x];
    for (int w = 0; w < NWAVES; ++w) {
      s_ct[w] = run;
      run = s_tany[w] ? s_tsum[w] : (run + s_tsum[w]);
    }
  }
  __syncthreads();

  // 4. final: segmented inclusive S, weights, per-ray totals at segment ends
  float ct = s_ct[wave];
  int a2 = ea; float px = ep;
  #pragma unroll
  for (int j = 0; j < IT; ++j) {
    int k  = ebase + j;
    int gi = cbase + k;
    float v  = s_v[k];
    float Pk = s_p[k];
    if (s_f[k]) { a2 = 1; px = Pk - v; }
    float S  = a2 ? (Pk - px) : (Pk + ct);        // segmented inclusive cumsum
    float ex = S - v;                             // exclusive (log T)
    float alpha = -expm1f(v);
    out_w[gi] = expf(ex) * alpha;
    int isLast;
    if (k < CHUNK - 1) isLast = (int)s_f[k + 1];
    else isLast = (gi == M_TOTAL - 1) ? 1 : (ray_id[gi + 1] != ray_id[gi]);
    if (isLast) out_a[ray_id[gi]] = expf(S);      // alphainv_last = exp(total)
  }
}

// ---------------------------------------------------------------------------
extern "C" void kernel_launch(void* const* d_in, const int* in_sizes, int n_in,
                              void* d_out, int out_size, void* d_ws, size_t ws_size,
                              hipStream_t stream) {
  (void)in_sizes; (void)n_in; (void)out_size; (void)ws_size;
  const float* density   = (const float*)d_in[0];
  const int*   ray_id    = (const int*)d_in[1];
  const float* shiftp    = (const float*)d_in[2];
  const float* intervalp = (const float*)d_in[3];

  float* out_w = (float*)d_out;                 // weights [M]
  float* out_a = out_w + M_TOTAL;               // alphainv_last [N]

  float* carrySum = (float*)d_ws;                                   // [NBLK]
  int*   carryFlag = (int*)((char*)d_ws + NBLK * sizeof(float));    // [NBLK]
  float* carryIn  = (float*)((char*)d_ws + 2 * NBLK * sizeof(float)); // [NBLK]

  k_init<<<(NRAYS + BT - 1) / BT, BT, 0, stream>>>(out_a);
  k_blockcarry<<<NBLK, BT, 0, stream>>>(density, ray_id, shiftp, intervalp,
                                        carrySum, carryFlag);
  k_scancarries<<<1, BT, 0, stream>>>(carrySum, carryFlag, carryIn);
  k_main<<<NBLK, BT, 0, stream>>>(density, ray_id, shiftp, intervalp, carryIn,
                                  out_w, out_a);
}